// EWRLSLevel_14396730377049
// MI455X (gfx1250) — compile-verified
//
#include <hip/hip_runtime.h>
#include <cstdint>
#include <cstddef>

// Problem constants (from reference): B=64, T=1024, C=512, float32.
#define B_DIM   64
#define T_DIM   1024
#define C_DIM   512

#define THREADS 32                  // one wave32 per block, barrier-free
#define CPT     4                   // channels per thread (float4 / B128)
#define CSLAB   (THREADS * CPT)     // 128 channels per block
#define SLABS   (C_DIM / CSLAB)     // 4 channel slabs
#define CHUNK   8                   // t-rows staged per chunk
#define NCHUNK  (T_DIM / CHUNK)     // 128 chunks
#define PIPE    7                   // chunks in flight (7*8 = 56 async ops <= 63)
#define NBUF    8                   // LDS ring slots (power of two, >= PIPE+1)
#define SLOT_FLOATS (CHUNK * CSLAB)             // 1024 floats = 4 KB per slot
#define LDS_MAIN_FLOATS (NBUF * SLOT_FLOATS)    // 8192 floats = 32 KB
#define LDS_TOTAL_BYTES (LDS_MAIN_FLOATS * 4 + THREADS * 16)  // + dummy scratch

typedef int v4i __attribute__((ext_vector_type(4)));

// ---- CDNA5-specific ops: builtin if present, inline asm fallback ----------
#if defined(__has_builtin)
#  if __has_builtin(__builtin_amdgcn_global_load_async_to_lds_b128)
#    define HAVE_ASYNC_B128 1
#  endif
#  if __has_builtin(__builtin_amdgcn_s_wait_asynccnt)
#    define HAVE_WAIT_ASYNC 1
#  endif
#endif

__device__ __forceinline__ void async_row_b128(const float* g, float* l) {
#if defined(HAVE_ASYNC_B128)
  // Probe-confirmed param types: pointer-to-v4i (AS1 source, AS3 LDS dest).
  __builtin_amdgcn_global_load_async_to_lds_b128(
      (__attribute__((address_space(1))) v4i*)(float*)g,
      (__attribute__((address_space(3))) v4i*)l,
      /*imm offset*/ 0, /*cpol*/ 0);
#else
  unsigned loff = (unsigned)(size_t)(__attribute__((address_space(3))) void*)l;
  asm volatile("global_load_async_to_lds_b128 %0, %1, off"
               :: "v"(loff), "v"(g) : "memory");
#endif
}

template <int N>
__device__ __forceinline__ void wait_asynccnt_le() {
#if defined(HAVE_WAIT_ASYNC)
  __builtin_amdgcn_s_wait_asynccnt(N);
#else
  asm volatile("s_wait_asynccnt %0" :: "i"(N) : "memory");
#endif
  asm volatile("" ::: "memory");   // compiler ordering fence for LDS reads
}

// Issue one 8-row chunk: per lane, 8 async B128 transfers (global row stride
// 2048 B, LDS row stride 512 B). One wave instruction moves 512 contiguous B.
__device__ __forceinline__ void issue_chunk(const float* gbase, float* smem,
                                            int n, int tid) {
  const float* g = gbase + (size_t)n * CHUNK * C_DIM;
  float* l = smem + (size_t)(n & (NBUF - 1)) * SLOT_FLOATS + tid * CPT;
#pragma unroll
  for (int r = 0; r < CHUNK; ++r) {
    async_row_b128(g, l);
    g += C_DIM;    // next t-row in global
    l += CSLAB;    // next row in LDS slot
  }
}

// Dummy chunk: keeps ASYNCcnt bookkeeping uniform for the tail of the pipe.
// Reads valid addresses (row 0) into a scratch LDS region nobody consumes.
__device__ __forceinline__ void issue_dummy(const float* gbase, float* dummy_l) {
#pragma unroll
  for (int r = 0; r < CHUNK; ++r) async_row_b128(gbase, dummy_l);
}

__global__ void __launch_bounds__(THREADS, 1)
ewrls_level_kernel(const float* __restrict__ x,
                   const float* __restrict__ logit_lambda,
                   float* __restrict__ y) {
  extern __shared__ __align__(16) float smem[];

  const int tid  = threadIdx.x;
  const int blk  = blockIdx.x;        // 0..B_DIM*SLABS-1
  const int b    = blk >> 2;          // / SLABS
  const int slab = blk & (SLABS - 1);
  const int c0   = slab * CSLAB + tid * CPT;

  // lam = clip(sigmoid(logit_lambda), 1e-4, 1-1e-4); per-lane float4 slab.
  float lam[CPT], r[CPT], th[CPT];
  const float4 lz = *reinterpret_cast<const float4*>(logit_lambda + c0);
  const float lzv[CPT] = {lz.x, lz.y, lz.z, lz.w};
#pragma unroll
  for (int i = 0; i < CPT; ++i) {
    float s = 1.0f / (1.0f + __expf(-lzv[i]));
    s = fminf(fmaxf(s, 1.0e-4f), 1.0f - 1.0e-4f);
    lam[i] = s;
    r[i]   = 1.0f;   // r_t = sum_{i<=t} lam^i ; K_t = 1/r_t (== P_t)
  }

  const float* gbase = x + (size_t)b * T_DIM * C_DIM + c0;
  float*       obase = y + (size_t)b * T_DIM * C_DIM + c0;
  float* dummy_l = smem + LDS_MAIN_FLOATS + tid * 4;

  // Prologue: fill the pipeline with PIPE chunks (56 outstanding async ops).
#pragma unroll
  for (int k = 0; k < PIPE; ++k) issue_chunk(gbase, smem, k, tid);

  for (int n = 0; n < NCHUNK; ++n) {
    if (n + PIPE < NCHUNK) issue_chunk(gbase, smem, n + PIPE, tid);
    else                   issue_dummy(gbase, dummy_l);

    // Chunks after n contribute exactly PIPE*CHUNK outstanding ops; async
    // loads complete in order, so <= PIPE*CHUNK implies chunk n has landed.
    wait_asynccnt_le<PIPE * CHUNK>();

    const float* lrow = smem + (size_t)(n & (NBUF - 1)) * SLOT_FLOATS + tid * CPT;
    float*       orow = obase + (size_t)n * CHUNK * C_DIM;
#pragma unroll
    for (int rr = 0; rr < CHUNK; ++rr) {
      const float4 xv = *reinterpret_cast<const float4*>(lrow);
      const float xs[CPT] = {xv.x, xv.y, xv.z, xv.w};
      if (n == 0 && rr == 0) {
        // t = 0: theta0 = x[:,0,:], y[:,0,:] = theta0
#pragma unroll
        for (int i = 0; i < CPT; ++i) th[i] = xs[i];
      } else {
#pragma unroll
        for (int i = 0; i < CPT; ++i) {
          r[i] = __builtin_fmaf(r[i], lam[i], 1.0f);       // r_t = lam*r + 1
          const float K = __builtin_amdgcn_rcpf(r[i]);      // K_t = 1/r_t
          th[i] = __builtin_fmaf(K, xs[i] - th[i], th[i]);  // theta += K*(x-theta)
        }
      }
      float4 ov; ov.x = th[0]; ov.y = th[1]; ov.z = th[2]; ov.w = th[3];
      *reinterpret_cast<float4*>(orow) = ov;
      lrow += CSLAB;
      orow += C_DIM;
    }
  }
}

extern "C" void kernel_launch(void* const* d_in, const int* in_sizes, int n_in,
                              void* d_out, int out_size, void* d_ws, size_t ws_size,
                              hipStream_t stream) {
  (void)in_sizes; (void)n_in; (void)out_size; (void)d_ws; (void)ws_size;
  const float* x  = (const float*)d_in[0];
  const float* lz = (const float*)d_in[1];
  float*       y  = (float*)d_out;

  dim3 grid(B_DIM * SLABS);   // 256 blocks
  dim3 block(THREADS);        // 1 wave32 each
  ewrls_level_kernel<<<grid, block, LDS_TOTAL_BYTES, stream>>>(x, lz, y);
}